// FastfoodProjection_27642409517621
// MI455X (gfx1250) — compile-verified
//
#include <hip/hip_runtime.h>
#include <hip/hip_bf16.h>
#include <math.h>

// Fastfood projection, D = 2^24, d = 2^13.
//
// out = theta + c * FWHT_unnorm_D( B[j] * w[perm[j] & 8191] )
//   w = FWHT_unnorm_8192(z * H),  c = 1/(D*sqrt(d)) folded into w.
//
// FWHT_D done in two LDS-staged passes; 4 stages at a time via
// V_WMMA_F32_16X16X4_F32 (multiply by the 16x16 Hadamard matrix H16).
// Single-use streams (perm, B, theta, final out) use non-temporal hints so
// the 64MB intermediate stays resident in the 192MB L2 between passes.

#define D_TOTAL   (1 << 24)
#define D_SMALL   8192
#define CHUNK     65536           // 2^16 elements per workgroup in pass 1
#define T3_ROWS   256             // pass-2 rows (stride 65536)
#define T3_COLS   64              // pass-2 contiguous columns

typedef __attribute__((ext_vector_type(2))) float v2f;
typedef __attribute__((ext_vector_type(4))) float f4;
typedef __attribute__((ext_vector_type(8))) float v8f;
typedef __attribute__((ext_vector_type(4))) int   i4;

// Build H16 tiles in WMMA f32 A-matrix layout:
//   lane l: m = l&15, hi = l>>4;  A[q] covers K = 4q .. 4q+3
//   A[q].x -> K = 4q + 2*hi,  A[q].y -> K = 4q + 1 + 2*hi
__device__ __forceinline__ void make_h16_a(int m, int hi, v2f A[4]) {
#pragma unroll
  for (int q = 0; q < 4; ++q) {
    int k0 = 4 * q + 2 * hi;
    A[q].x = (__popc(m & k0) & 1) ? -1.0f : 1.0f;
    A[q].y = (__popc(m & (k0 + 1)) & 1) ? -1.0f : 1.0f;
  }
}

// Apply Y = H16 * X in place on an LDS tile.
// Element X[k][n] lives at s[base + k*S + n*T]; one wave handles one tile.
// B layout: lane l supplies (K = 4q+2*hi [+1], N = m).
// D layout: lane l holds rows r+8*hi (r=0..7), col m.
__device__ __forceinline__ void h16_apply(float* s, int base, int S, int T,
                                          const v2f A[4], int m, int hi) {
  v8f acc = {0.f, 0.f, 0.f, 0.f, 0.f, 0.f, 0.f, 0.f};
#pragma unroll
  for (int q = 0; q < 4; ++q) {
    int k0 = 4 * q + 2 * hi;
    v2f b;
    b.x = s[base + k0 * S + m * T];
    b.y = s[base + (k0 + 1) * S + m * T];
    acc = __builtin_amdgcn_wmma_f32_16x16x4_f32(
        /*neg_a=*/false, A[q], /*neg_b=*/false, b,
        /*c_mod=*/(short)0, acc, /*reuse_a=*/false, /*reuse_b=*/false);
  }
#pragma unroll
  for (int r = 0; r < 8; ++r) {
    s[base + (r + 8 * hi) * S + m * T] = acc[r];
  }
}

// ---------------- Kernel 1: w = c * FWHT_unnorm_8192(z * H) ----------------
__global__ __launch_bounds__(1024) void k_small_fwht(
    const float* __restrict__ z, const float* __restrict__ H,
    float* __restrict__ w) {
  __shared__ float s[D_SMALL];
  const int t = threadIdx.x;
  const float c = (float)(1.0 / (16777216.0 * 90.50966799187809));  // 1/(D*sqrt(d))
#pragma unroll
  for (int i = 0; i < D_SMALL / 1024; ++i) {
    int k = t + i * 1024;
    s[k] = z[k] * H[k] * c;
  }
  __syncthreads();
  for (int st = 0; st < 13; ++st) {
    int h = 1 << st;
#pragma unroll
    for (int pi = 0; pi < (D_SMALL / 2) / 1024; ++pi) {
      int p = t + pi * 1024;
      int idx = ((p >> st) << (st + 1)) | (p & (h - 1));
      float a = s[idx];
      float b = s[idx + h];
      s[idx] = a + b;
      s[idx + h] = a - b;
    }
    __syncthreads();
  }
#pragma unroll
  for (int i = 0; i < D_SMALL / 1024; ++i) {
    int k = t + i * 1024;
    w[k] = s[k];
  }
}

// ------- Kernel 2: gather + FWHT stages 0..15 on contiguous chunks ---------
__global__ __launch_bounds__(512) void k_fwht_pass1(
    const float* __restrict__ w, const int* __restrict__ perm,
    const float* __restrict__ Bsgn, float* __restrict__ out) {
  __shared__ float s[CHUNK];  // 256 KB
  const int t = threadIdx.x;
  const long long gbase = (long long)blockIdx.x * CHUNK;

  // Gather: s[f] = B[j] * w[perm[j] & 8191], j = gbase + f.
  // perm/B are single-use streams -> non-temporal; w (32KB) stays cache-hot.
#pragma unroll 4
  for (int i = 0; i < CHUNK / (512 * 4); ++i) {
    int f = 4 * (t + i * 512);
    i4 p  = __builtin_nontemporal_load((const i4*)(perm + gbase + f));
    f4 bb = __builtin_nontemporal_load((const f4*)(Bsgn + gbase + f));
    f4 y;
    y.x = bb.x * w[p.x & (D_SMALL - 1)];
    y.y = bb.y * w[p.y & (D_SMALL - 1)];
    y.z = bb.z * w[p.z & (D_SMALL - 1)];
    y.w = bb.w * w[p.w & (D_SMALL - 1)];
    *(f4*)(s + f) = y;
  }
  __syncthreads();

  const int lane = t & 31;
  const int wave = __builtin_amdgcn_readfirstlane(t >> 5);  // scalar wave id
  const int m = lane & 15, hi = lane >> 4;
  v2f A[4];
  make_h16_a(m, hi, A);

  // Chunk viewed as 16^4: index = a*4096 + b*256 + cdig*16 + e.
  // Apply H16 along each hex digit; 256 disjoint 16x16 tiles per axis,
  // 16 tiles per wave, scalar counted loops (EXEC stays all-ones).
#pragma unroll 2
  for (int i = 0; i < 16; ++i) {            // axis e: S=1, n over cdig (T=16)
    int tt = (i << 4) | wave;
    h16_apply(s, tt * 256, 1, 16, A, m, hi);
  }
  __syncthreads();
#pragma unroll 2
  for (int i = 0; i < 16; ++i) {            // axis cdig: S=16, n over e
    int tt = (i << 4) | wave;
    h16_apply(s, tt * 256, 16, 1, A, m, hi);
  }
  __syncthreads();
#pragma unroll 2
  for (int i = 0; i < 16; ++i) {            // axis b: S=256, n over e
    int tt = (i << 4) | wave;
    h16_apply(s, (tt & 15) * 16 + (tt >> 4) * 4096, 256, 1, A, m, hi);
  }
  __syncthreads();
#pragma unroll 2
  for (int i = 0; i < 16; ++i) {            // axis a: S=4096, n over e
    int tt = (i << 4) | wave;
    h16_apply(s, tt * 16, 4096, 1, A, m, hi);
  }
  __syncthreads();

  // Intermediate to d_out: re-read by pass 2 -> keep temporal (L2-resident).
#pragma unroll 4
  for (int i = 0; i < CHUNK / (512 * 4); ++i) {
    int f = 4 * (t + i * 512);
    *(f4*)(out + gbase + f) = *(const f4*)(s + f);
  }
}

// ------ Kernel 3: FWHT stages 16..23 (stride 65536) + theta, in place ------
__global__ __launch_bounds__(512) void k_fwht_pass2(
    const float* __restrict__ theta, float* __restrict__ out) {
  __shared__ float s[T3_ROWS * T3_COLS];  // 64 KB; s[h*64 + c]
  const int t = threadIdx.x;
  const long long cb = (long long)blockIdx.x * T3_COLS;

  // Load 256 rows (global stride 65536) x 64 contiguous cols: 256B per row.
  // Intermediate should be L2-resident from pass 1 -> default temporal load.
#pragma unroll
  for (int i = 0; i < (T3_ROWS * T3_COLS) / (512 * 4); ++i) {  // 8 iters
    int f = 4 * (t + i * 512);
    int h = f >> 6, c = f & 63;
    *(f4*)(s + f) = *(const f4*)(out + (long long)h * CHUNK + cb + c);
  }
  __syncthreads();

  const int lane = t & 31;
  const int wave = __builtin_amdgcn_readfirstlane(t >> 5);
  const int m = lane & 15, hi = lane >> 4;
  v2f A[4];
  make_h16_a(m, hi, A);

  // h = h1*16 + h0; LDS stride of h is 64. 4 tiles per wave per axis.
#pragma unroll 2
  for (int i = 0; i < 4; ++i) {             // axis h0: S=64
    int tt = (i << 4) | wave;
    int h1 = tt >> 2, cg = tt & 3;
    h16_apply(s, h1 * 1024 + cg * 16, 64, 1, A, m, hi);
  }
  __syncthreads();
#pragma unroll 2
  for (int i = 0; i < 4; ++i) {             // axis h1: S=1024
    int tt = (i << 4) | wave;
    int h0 = tt >> 2, cg = tt & 3;
    h16_apply(s, h0 * 64 + cg * 16, 1024, 1, A, m, hi);
  }
  __syncthreads();

  // out = theta + x. theta is single-use and the result is never re-read
  // -> non-temporal both ways.
#pragma unroll
  for (int i = 0; i < (T3_ROWS * T3_COLS) / (512 * 4); ++i) {
    int f = 4 * (t + i * 512);
    int h = f >> 6, c = f & 63;
    long long gi = (long long)h * CHUNK + cb + c;
    f4 th = __builtin_nontemporal_load((const f4*)(theta + gi));
    f4 v = *(const f4*)(s + f);
    v.x += th.x; v.y += th.y; v.z += th.z; v.w += th.w;
    __builtin_nontemporal_store(v, (f4*)(out + gi));
  }
}

extern "C" void kernel_launch(void* const* d_in, const int* in_sizes, int n_in,
                              void* d_out, int out_size, void* d_ws, size_t ws_size,
                              hipStream_t stream) {
  // setup_inputs order: z, theta_base, perm, B, H
  const float* z     = (const float*)d_in[0];
  const float* theta = (const float*)d_in[1];
  const int*   perm  = (const int*)d_in[2];
  const float* Bsgn  = (const float*)d_in[3];
  const float* H     = (const float*)d_in[4];
  float* out = (float*)d_out;
  float* w   = (float*)d_ws;  // 8192 floats = 32 KB scratch

  k_small_fwht<<<1, 1024, 0, stream>>>(z, H, w);
  k_fwht_pass1<<<D_TOTAL / CHUNK, 512, 0, stream>>>(w, perm, Bsgn, out);
  k_fwht_pass2<<<CHUNK / T3_COLS, 512, 0, stream>>>(theta, out);
}